// ESNReservoir_74998718923292
// MI455X (gfx1250) — compile-verified
//
#include <hip/hip_runtime.h>

typedef __attribute__((ext_vector_type(2))) float v2f;
typedef __attribute__((ext_vector_type(4))) float v4f;
typedef __attribute__((ext_vector_type(8))) float v8f;
typedef __attribute__((ext_vector_type(4))) int   v4i;

#define T_STEPS 8192
#define F_IN    64
#define R_RES   2048
#define LEAK    0.9f

#define NWG     128     // persistent blocks; 16 output rows each
#define BLOCK   256     // 8 wave32s; each wave owns K-slice of 2048/8 = 256
#define ROWS    16      // output rows per block

// -------- CDNA5 async global->LDS copies (guarded; fallback = plain copy) ---
#if defined(__has_builtin)
#if __has_builtin(__builtin_amdgcn_global_load_async_to_lds_b32) && \
    __has_builtin(__builtin_amdgcn_global_load_async_to_lds_b128) && \
    __has_builtin(__builtin_amdgcn_s_wait_asynccnt)
#define HAVE_ASYNC_LDS 1
#endif
#endif

#if defined(HAVE_ASYNC_LDS)
using g_i32 = __attribute__((address_space(1))) int;
using l_i32 = __attribute__((address_space(3))) int;
using g_v4i = __attribute__((address_space(1))) v4i;
using l_v4i = __attribute__((address_space(3))) v4i;
#define ASYNC_B32(g, l)  __builtin_amdgcn_global_load_async_to_lds_b32( \
    (g_i32*)(g), (l_i32*)(l), 0, 0)
#define ASYNC_B128(g, l) __builtin_amdgcn_global_load_async_to_lds_b128( \
    (g_v4i*)(g), (l_v4i*)(l), 0, 0)
#define ASYNC_WAIT()     __builtin_amdgcn_s_wait_asynccnt(0)
#endif

// ---------------------------------------------------------------------------
// Grid-wide barrier: alternating arrival counters + generation word in d_ws.
// All x-state writes are done by wave 0 (threads 0..15), the same wave that
// executes the ACQ_REL arrive, so store ordering is wave-local and airtight.
// ---------------------------------------------------------------------------
__device__ __forceinline__ void grid_barrier(unsigned* bar, unsigned bi) {
  __syncthreads();
  if (threadIdx.x == 0) {
    unsigned* cnt = bar + ((bi & 1u) ? 32 : 0);
    unsigned* gen = bar + 64;
    unsigned prev = __hip_atomic_fetch_add(cnt, 1u, __ATOMIC_ACQ_REL,
                                           __HIP_MEMORY_SCOPE_AGENT);
    if (prev == (unsigned)(NWG - 1)) {
      __hip_atomic_store(cnt, 0u, __ATOMIC_RELAXED, __HIP_MEMORY_SCOPE_AGENT);
      __hip_atomic_store(gen, bi + 1u, __ATOMIC_RELEASE, __HIP_MEMORY_SCOPE_AGENT);
    } else {
      while (__hip_atomic_load(gen, __ATOMIC_ACQUIRE, __HIP_MEMORY_SCOPE_AGENT) <= bi) {
        __builtin_amdgcn_s_sleep(2);
      }
    }
  }
  __syncthreads();
  __threadfence();   // refresh every CU's L0 view of x[t]
}

// ---------------------------------------------------------------------------
// Kernel 1: proj[t][r] = sum_k u[t][k]*Win[r][k] + b[r]   (fp32 WMMA GEMM)
// One wave per 16x16 tile; K=64 in 16 V_WMMA_F32_16X16X4_F32 steps.
// Operand layout per ISA 7.12.2: lanes 0-15 K={0,1}, lanes 16-31 K={2,3}.
// ---------------------------------------------------------------------------
__global__ __launch_bounds__(BLOCK) void esn_proj_wmma(
    const float* __restrict__ u, const float* __restrict__ Win,
    const float* __restrict__ bias, float* __restrict__ proj)
{
  const int lane  = threadIdx.x & 31;
  const int wave  = threadIdx.x >> 5;
  const int t0    = blockIdx.x * 16;
  const int r0    = (blockIdx.y * 8 + wave) * 16;
  const int mrow  = lane & 15;
  const int khalf = (lane >> 4) * 2;

  v8f c;
  {
    float bv = bias[r0 + mrow];
    #pragma unroll
    for (int j = 0; j < 8; ++j) c[j] = bv;
  }

  const float* aptr = u   + (t0 + mrow) * F_IN + khalf;
  const float* bptr = Win + (r0 + mrow) * F_IN + khalf;
  #pragma unroll
  for (int k0 = 0; k0 < F_IN; k0 += 4) {
    v2f a  = *(const v2f*)aptr;  aptr += 4;
    v2f bb = *(const v2f*)bptr;  bptr += 4;
    c = __builtin_amdgcn_wmma_f32_16x16x4_f32(false, a, false, bb,
                                              (short)0, c, false, false);
  }

  const int col     = r0 + mrow;
  const int rowbase = t0 + ((lane >> 4) * 8);
  #pragma unroll
  for (int j = 0; j < 8; ++j)
    proj[(size_t)(rowbase + j) * R_RES + col] = c[j];
}

// ---------------------------------------------------------------------------
// Kernel 2: persistent scan, Wres slice resident in LDS.
// Block b owns rows [16b,16b+16): 16*2048*4B = 128KB LDS, filled once via
// async global->LDS. Per step: stage x[t-1] (8KB) into LDS async, then each
// wave runs 64 WMMAs with A (x, broadcast across M) and B (Wres tile) both
// fed by ds_load_b64. Partial y lands in D VGPR0 lanes 0-15.
// ---------------------------------------------------------------------------
__global__ __launch_bounds__(BLOCK) void esn_scan_wmma(
    const float* __restrict__ W, float* __restrict__ X,
    unsigned* __restrict__ bar)
{
  __shared__ float lds_w[ROWS * R_RES];   // 128 KB
  __shared__ float lds_x[R_RES];          // 8 KB
  __shared__ float red[8][16];

  const int lane  = threadIdx.x & 31;
  const int wave  = threadIdx.x >> 5;
  const int n0    = blockIdx.x * ROWS;
  const int mrow  = lane & 15;
  const int khalf = (lane >> 4) * 2;
  const int kbase = wave * 256 + khalf;

  // ---- one-time fill: Wres rows [n0, n0+16) -> LDS (16B per thread/pass) ---
  {
    const float* wslice = W + (size_t)n0 * R_RES;
    #pragma unroll 4
    for (int p = 0; p < (ROWS * R_RES) / (BLOCK * 4); ++p) {
      int idx = (p * BLOCK + threadIdx.x) * 4;          // float index, 16B units
#if defined(HAVE_ASYNC_LDS)
      ASYNC_B128(wslice + idx, &lds_w[idx]);
#else
      *(v4f*)&lds_w[idx] = *(const v4f*)&wslice[idx];
#endif
    }
#if defined(HAVE_ASYNC_LDS)
    ASYNC_WAIT();
#endif
  }

  // ---- step 0: x0 = tanh(proj[0]) ----
  if (threadIdx.x < 16) {
    float p = X[n0 + threadIdx.x];
    X[n0 + threadIdx.x] = tanhf(p);
  }
  grid_barrier(bar, 0u);

  const float* wlds = &lds_w[mrow * R_RES + kbase];   // B operand base (LDS)

  for (int t = 1; t < T_STEPS; ++t) {
    const float* xprev = X + (size_t)(t - 1) * R_RES;

    // stage full x[t-1] into LDS (8 words per thread, coalesced)
    #pragma unroll
    for (int j = 0; j < R_RES / BLOCK; ++j) {
      int idx = j * BLOCK + threadIdx.x;
#if defined(HAVE_ASYNC_LDS)
      ASYNC_B32(xprev + idx, &lds_x[idx]);
#else
      lds_x[idx] = xprev[idx];
#endif
    }
#if defined(HAVE_ASYNC_LDS)
    ASYNC_WAIT();
#endif
    __syncthreads();

    v8f c = {};
    const float* xp = &lds_x[kbase];
    const float* wp = wlds;
    #pragma unroll 4
    for (int i = 0; i < 64; ++i) {
      v2f a  = *(const v2f*)xp;  xp += 4;   // x chunk (lanes 0-15 broadcast)
      v2f bb = *(const v2f*)wp;  wp += 4;   // Wres tile from LDS
      c = __builtin_amdgcn_wmma_f32_16x16x4_f32(false, a, false, bb,
                                                (short)0, c, false, false);
    }

    if (lane < 16) red[wave][lane] = c[0];
    __syncthreads();

    if (threadIdx.x < 16) {
      float y = 0.0f;
      #pragma unroll
      for (int wv = 0; wv < 8; ++wv) y += red[wv][threadIdx.x];
      const size_t o = (size_t)t * R_RES + n0 + threadIdx.x;
      float p   = X[o];                        // proj[t]
      float xpv = lds_x[n0 + threadIdx.x];     // x[t-1] (already staged)
      float xi  = tanhf(p + y);
      X[o] = (1.0f - LEAK) * xpv + LEAK * xi;
    }

    grid_barrier(bar, (unsigned)t);            // also protects lds_x reuse
  }
}

extern "C" void kernel_launch(void* const* d_in, const int* in_sizes, int n_in,
                              void* d_out, int out_size, void* d_ws, size_t ws_size,
                              hipStream_t stream) {
  const float* u     = (const float*)d_in[0];   // [T, F]
  const float* Win_w = (const float*)d_in[1];   // [R, F]
  const float* Win_b = (const float*)d_in[2];   // [R]
  const float* Wres  = (const float*)d_in[3];   // [R, R]
  float*       X     = (float*)d_out;           // [T, R], proj -> x in place
  unsigned*    bar   = (unsigned*)d_ws;

  size_t clr = ws_size < 320 ? ws_size : 320;
  (void)hipMemsetAsync(d_ws, 0, clr, stream);   // reset barrier state each call

  dim3 g1(T_STEPS / 16, R_RES / (16 * 8));      // (512, 16)
  esn_proj_wmma<<<g1, BLOCK, 0, stream>>>(u, Win_w, Win_b, X);

  esn_scan_wmma<<<NWG, BLOCK, 0, stream>>>(Wres, X, bar);
}